// DSAM_51694226375082
// MI455X (gfx1250) — compile-verified
//
#include <hip/hip_runtime.h>
#include <hip/hip_bf16.h>
#include <math.h>

// ---------------------------------------------------------------------------
// DANet dual-attention module for MI455X (gfx1250), all-f32 via
// V_WMMA_F32_16X16X4_F32. Flash-style spatial attention (A never
// materialized -> avoids 134MB of HBM traffic). wave32, 8 waves/block.
// V chunks staged with CDNA5 async global->LDS DMA (ASYNCcnt).
// ---------------------------------------------------------------------------

typedef float v2f __attribute__((ext_vector_type(2)));
typedef float v8f __attribute__((ext_vector_type(8)));

constexpr int CH  = 64;     // channels
constexpr int HT  = 64;     // height
constexpr int WD  = 64;     // width
constexpr int HWP = 4096;   // H*W
constexpr int K9  = 576;    // C*3*3 (im2col K for 3x3 conv)

__device__ __forceinline__ v8f wmma_f32(v2f a, v2f b, v8f c) {
  // D = A(16x4) * B(4x16) + C(16x16), f32 everywhere.
  return __builtin_amdgcn_wmma_f32_16x16x4_f32(
      /*neg_a=*/false, a, /*neg_b=*/false, b,
      /*c_mod=*/(short)0, c, /*reuse_a=*/false, /*reuse_b=*/false);
}

// LDS byte offset of a shared-memory object (generic -> addrspace(3) -> int).
__device__ __forceinline__ unsigned lds_offset(const void* p) {
  return (unsigned)(unsigned long long)
      (__attribute__((address_space(3))) const char*)(p);
}

// CDNA5 async global->LDS copy, 16 bytes per lane. Tracked by ASYNCcnt.
__device__ __forceinline__ void async_copy_b128(unsigned lds_off,
                                                const float* gsrc) {
  asm volatile("global_load_async_to_lds_b128 %0, %1, off"
               :: "v"(lds_off), "v"(gsrc)
               : "memory");
}
__device__ __forceinline__ void wait_async0() {
  asm volatile("s_wait_asynccnt 0x0" ::: "memory");
}

// ---------------------------------------------------------------------------
// Kernel 1: 3x3 conv (pad=1) + eval BatchNorm + ReLU, implicit GEMM.
// grid = (H, B). One block computes all 64 out-channels for one image row.
// M=64 (out ch), N=64 (pixels of the row), K=576.
// K-order is (dy,dx)-major / channel-minor so the 3x3 tap is wave-uniform
// per 64-wide K block: no per-lane div/mod in the hot loop.
// ---------------------------------------------------------------------------
__global__ __launch_bounds__(256)
void conv3x3_bn_relu(const float* __restrict__ x, const float* __restrict__ w,
                     const float* __restrict__ bias,
                     const float* __restrict__ bng, const float* __restrict__ bnb,
                     const float* __restrict__ bnm, const float* __restrict__ bnv,
                     float* __restrict__ out) {
  const int y = blockIdx.x;
  const int b = blockIdx.y;
  __shared__ float patch[3][CH][WD + 2];  // [dy][ch][x+1], 50688 B

  const float* xb = x + (size_t)b * CH * HWP;
  for (int idx = threadIdx.x; idx < 3 * CH * (WD + 2); idx += 256) {
    int col = idx % (WD + 2);
    int rest = idx / (WD + 2);
    int ch = rest % CH;
    int dy = rest / CH;
    int yy = y + dy - 1;
    int xx = col - 1;
    float val = 0.f;
    if (yy >= 0 && yy < HT && xx >= 0 && xx < WD)
      val = xb[ch * HWP + yy * WD + xx];
    patch[dy][ch][col] = val;
  }
  __syncthreads();

  const int lane = threadIdx.x & 31;
  const int wv   = threadIdx.x >> 5;
  const int lo   = lane & 15;
  const int hi   = lane >> 4;
  const int nt   = wv & 3;           // pixel tile (16 px)
  const int m0   = (wv >> 2) * 32;   // 2 out-channel tiles: m0, m0+16

  const float* wrow0 = w + (size_t)(m0 + lo) * K9;       // weights row, tile 0
  const float* wrow1 = wrow0 + 16 * K9;                  // tile 1

  v8f acc0 = {}; v8f acc1 = {};
  for (int t9 = 0; t9 < 9; ++t9) {      // 3x3 tap (wave-uniform)
    const int dy = t9 / 3;
    const int dx = t9 - dy * 3;
    const float* prow = &patch[dy][0][nt * 16 + lo + dx];
    for (int kc = 0; kc < CH; kc += 4) {  // channel chunk of 4
      const int ch = kc + hi * 2;
      v2f bf, a0, a1;
      bf[0] = prow[(size_t)ch * (WD + 2)];
      bf[1] = prow[(size_t)(ch + 1) * (WD + 2)];
      // weight K index = ch*9 + t9 (OIHW layout)
      a0[0] = wrow0[ch * 9 + t9];
      a0[1] = wrow0[ch * 9 + 9 + t9];
      a1[0] = wrow1[ch * 9 + t9];
      a1[1] = wrow1[ch * 9 + 9 + t9];
      acc0 = wmma_f32(a0, bf, acc0);
      acc1 = wmma_f32(a1, bf, acc1);
    }
  }

  const int px = nt * 16 + lo;
  float* ob = out + ((size_t)b * CH) * HWP + y * WD;
#pragma unroll
  for (int r = 0; r < 8; ++r) {
    int o = m0 + r + hi * 8;
    float sc = bng[o] * rsqrtf(bnv[o] + 1e-5f);
    float v0 = (acc0[r] + bias[o] - bnm[o]) * sc + bnb[o];
    ob[(size_t)o * HWP + px] = fmaxf(v0, 0.f);
    int o1 = o + 16;
    float sc1 = bng[o1] * rsqrtf(bnv[o1] + 1e-5f);
    float v1 = (acc1[r] + bias[o1] - bnm[o1]) * sc1 + bnb[o1];
    ob[(size_t)o1 * HWP + px] = fmaxf(v1, 0.f);
  }
}

// ---------------------------------------------------------------------------
// Kernel 2: 1x1 conv: out[M, HW] = W[M, C] * in[C, HW] + bias.  M = 8 or 64.
// grid = (HW/64, B).
// ---------------------------------------------------------------------------
__global__ __launch_bounds__(256)
void conv1x1(const float* __restrict__ x, const float* __restrict__ w,
             const float* __restrict__ bias, float* __restrict__ out, int M) {
  const int p0 = blockIdx.x * 64;
  const int b  = blockIdx.y;
  __shared__ float xt[CH][64 + 2];

  const float* xb = x + (size_t)b * CH * HWP;
  for (int idx = threadIdx.x; idx < CH * 64; idx += 256) {
    int ch = idx >> 6;
    int px = idx & 63;
    xt[ch][px] = xb[ch * HWP + p0 + px];
  }
  __syncthreads();

  const int lane = threadIdx.x & 31, wv = threadIdx.x >> 5;
  const int lo = lane & 15, hi = lane >> 4;
  const int nt = wv & 3;

  for (int mt = wv >> 2; mt * 16 < M; mt += 2) {  // wave-uniform loop
    v8f acc = {};
    for (int k0 = 0; k0 < CH; k0 += 4) {
      int kk = k0 + hi * 2;
      int m = mt * 16 + lo;
      v2f a, bf;
      a[0] = (m < M) ? w[m * CH + kk] : 0.f;
      a[1] = (m < M) ? w[m * CH + kk + 1] : 0.f;
      bf[0] = xt[kk][nt * 16 + lo];
      bf[1] = xt[kk + 1][nt * 16 + lo];
      acc = wmma_f32(a, bf, acc);
    }
#pragma unroll
    for (int r = 0; r < 8; ++r) {
      int o = mt * 16 + r + hi * 8;
      if (o < M)
        out[((size_t)b * M + o) * HWP + p0 + nt * 16 + lo] = acc[r] + bias[o];
    }
  }
}

// ---------------------------------------------------------------------------
// Kernel 3: flash-style spatial attention + fuse with s_x.
//   samap[c,i] = gamma * ( sum_j softmax_j(q_i.k_j) * v[c,j] ) + s_x[c,i]
// grid = (HW/64, B). One block owns a 64-query block, streams 64-key chunks.
// V chunks arrive via GLOBAL_LOAD_ASYNC_TO_LDS_B128 overlapping the S-GEMM.
// ---------------------------------------------------------------------------
__global__ __launch_bounds__(256)
void spatial_attn(const float* __restrict__ q, const float* __restrict__ k,
                  const float* __restrict__ v, const float* __restrict__ sx,
                  const float* __restrict__ gamma, float* __restrict__ samap) {
  const int i0 = blockIdx.x * 64;
  const int b  = blockIdx.y;

  __shared__ float P[64][64 + 1];    // S scores, then exp(P)
  __shared__ float Vt[64][64 + 4];   // staged V chunk [c][j]; 272B row = 16B aligned
  __shared__ float red_m[4][64], red_s[4][64];
  __shared__ float rowm[64], rowl[64], alpha[64];

  const int tid = threadIdx.x;
  const int lane = tid & 31, wv = tid >> 5;
  const int lo = lane & 15, hi = lane >> 4;

  if (tid < 64) { rowm[tid] = -3.4e38f; rowl[tid] = 0.f; }

  const float* qb = q + (size_t)b * 8 * HWP;
  const float* kb = k + (size_t)b * 8 * HWP;
  const float* vb = v + (size_t)b * CH * HWP;

  // --- S-GEMM tile assignment: jtile = wv&3, i-tiles {2*(wv>>2), +1}
  const int s_jt = wv & 3;
  const int s_i0 = (wv >> 2) * 32;
  // Q A-fragments (invariant over j): A[m=i][k=d]
  v2f qa[2][2];
#pragma unroll
  for (int it = 0; it < 2; ++it)
#pragma unroll
    for (int ks = 0; ks < 2; ++ks) {
      qa[it][ks][0] = qb[(ks * 4 + hi * 2 + 0) * HWP + i0 + s_i0 + it * 16 + lo];
      qa[it][ks][1] = qb[(ks * 4 + hi * 2 + 1) * HWP + i0 + s_i0 + it * 16 + lo];
    }

  // --- O-GEMM tile assignment: i-tile = wv&3, c-tiles {2*(wv>>2), +1}
  const int o_it = wv & 3;
  const int o_c0 = (wv >> 2) * 32;
  v8f oacc0 = {}, oacc1 = {};
  __syncthreads();

  for (int j0 = 0; j0 < HWP; j0 += 64) {
    // stage V chunk via async DMA: 1024 x b128 = 16KB, overlaps S-GEMM below.
#pragma unroll
    for (int c = 0; c < 4; ++c) {
      int linear = tid + c * 256;          // 1024 16-byte chunks
      int ch = linear >> 4;                // 16 chunks per channel row
      int j4 = (linear & 15) * 4;
      async_copy_b128(lds_offset(&Vt[ch][j4]), vb + ch * HWP + j0 + j4);
    }
    if (j0 + 64 < HWP)
      __builtin_prefetch(kb + j0 + 64 + s_jt * 16, 0, 1);

    // S = Q^T K over K=8 (2 wmma steps), 2 i-tiles per wave
    v8f s0 = {}, s1 = {};
#pragma unroll
    for (int ks = 0; ks < 2; ++ks) {
      v2f bf;
      bf[0] = kb[(ks * 4 + hi * 2 + 0) * HWP + j0 + s_jt * 16 + lo];
      bf[1] = kb[(ks * 4 + hi * 2 + 1) * HWP + j0 + s_jt * 16 + lo];
      s0 = wmma_f32(qa[0][ks], bf, s0);
      s1 = wmma_f32(qa[1][ks], bf, s1);
    }
#pragma unroll
    for (int r = 0; r < 8; ++r) {
      P[s_i0 + r + hi * 8][s_jt * 16 + lo]      = s0[r];
      P[s_i0 + 16 + r + hi * 8][s_jt * 16 + lo] = s1[r];
    }
    wait_async0();        // V chunk resident in LDS before the barrier
    __syncthreads();

    // online softmax: thread handles (row = tid&63, 16-col quarter = tid>>6)
    {
      const int row = tid & 63, qd = tid >> 6;
      float mx = -3.4e38f;
#pragma unroll
      for (int c = 0; c < 16; ++c) mx = fmaxf(mx, P[row][qd * 16 + c]);
      red_m[qd][row] = mx;
      __syncthreads();
      float m_blk = fmaxf(fmaxf(red_m[0][row], red_m[1][row]),
                          fmaxf(red_m[2][row], red_m[3][row]));
      float m_old = rowm[row];
      float m_new = fmaxf(m_old, m_blk);
      float a = __expf(m_old - m_new);
      float s = 0.f;
#pragma unroll
      for (int c = 0; c < 16; ++c) {
        float e = __expf(P[row][qd * 16 + c] - m_new);
        P[row][qd * 16 + c] = e;
        s += e;
      }
      red_s[qd][row] = s;
      __syncthreads();
      if (qd == 0) {
        rowm[row] = m_new;
        rowl[row] = rowl[row] * a +
                    red_s[0][row] + red_s[1][row] + red_s[2][row] + red_s[3][row];
        alpha[row] = a;
      }
      __syncthreads();
    }

    // rescale accumulators (per query column i) and O += V * P^T
    float av = alpha[o_it * 16 + lo];
#pragma unroll
    for (int r = 0; r < 8; ++r) { oacc0[r] *= av; oacc1[r] *= av; }
    for (int ks = 0; ks < 64; ks += 4) {
      v2f bf, a0, a1;
      // B[k=j][n=i] = P[i][j]
      bf[0] = P[o_it * 16 + lo][ks + hi * 2 + 0];
      bf[1] = P[o_it * 16 + lo][ks + hi * 2 + 1];
      a0[0] = Vt[o_c0 + lo][ks + hi * 2 + 0];
      a0[1] = Vt[o_c0 + lo][ks + hi * 2 + 1];
      a1[0] = Vt[o_c0 + 16 + lo][ks + hi * 2 + 0];
      a1[1] = Vt[o_c0 + 16 + lo][ks + hi * 2 + 1];
      oacc0 = wmma_f32(a0, bf, oacc0);
      oacc1 = wmma_f32(a1, bf, oacc1);
    }
    __syncthreads();  // protect P / Vt for next chunk
  }

  // epilogue: samap = gamma * O / l + s_x
  const float g = gamma[0];
  const float invl = 1.f / rowl[o_it * 16 + lo];
  const int px = i0 + o_it * 16 + lo;
#pragma unroll
  for (int r = 0; r < 8; ++r) {
    int c0 = o_c0 + r + hi * 8;
    int c1 = c0 + 16;
    samap[((size_t)b * CH + c0) * HWP + px] =
        g * oacc0[r] * invl + sx[((size_t)b * CH + c0) * HWP + px];
    samap[((size_t)b * CH + c1) * HWP + px] =
        g * oacc1[r] * invl + sx[((size_t)b * CH + c1) * HWP + px];
  }
}

// ---------------------------------------------------------------------------
// Kernel 4: channel gram (64x64 over K=4096) + softmax(rowmax - Ac).
// grid = (B). Stable form: softmax_j(M_i - Ac[i,j]) == exp(min_j Ac - Ac)/sum.
// ---------------------------------------------------------------------------
__global__ __launch_bounds__(256)
void chan_gram_softmax(const float* __restrict__ cx, float* __restrict__ Asoft) {
  const int b = blockIdx.x;
  __shared__ float Xc[CH][64 + 2];
  __shared__ float Ac[CH][CH + 1];
  __shared__ float red_a[4][64], red_b[4][64];

  const int tid = threadIdx.x;
  const int lane = tid & 31, wv = tid >> 5;
  const int lo = lane & 15, hi = lane >> 4;
  const int nt = wv & 3, m0 = (wv >> 2) * 32;
  const float* cb = cx + (size_t)b * CH * HWP;

  v8f acc0 = {}, acc1 = {};
  for (int k0 = 0; k0 < HWP; k0 += 64) {
    __syncthreads();
    for (int idx = tid; idx < CH * 64; idx += 256) {
      int ch = idx >> 6, kk = idx & 63;
      Xc[ch][kk] = cb[ch * HWP + k0 + kk];
    }
    __syncthreads();
    for (int ks = 0; ks < 64; ks += 4) {
      v2f a0, a1, bf;
      bf[0] = Xc[nt * 16 + lo][ks + hi * 2 + 0];
      bf[1] = Xc[nt * 16 + lo][ks + hi * 2 + 1];
      a0[0] = Xc[m0 + lo][ks + hi * 2 + 0];
      a0[1] = Xc[m0 + lo][ks + hi * 2 + 1];
      a1[0] = Xc[m0 + 16 + lo][ks + hi * 2 + 0];
      a1[1] = Xc[m0 + 16 + lo][ks + hi * 2 + 1];
      acc0 = wmma_f32(a0, bf, acc0);
      acc1 = wmma_f32(a1, bf, acc1);
    }
  }
#pragma unroll
  for (int r = 0; r < 8; ++r) {
    Ac[m0 + r + hi * 8][nt * 16 + lo]      = acc0[r];
    Ac[m0 + 16 + r + hi * 8][nt * 16 + lo] = acc1[r];
  }
  __syncthreads();

  const int row = tid & 63, qd = tid >> 6;
  float mn = 3.4e38f;
#pragma unroll
  for (int c2 = 0; c2 < 16; ++c2) mn = fminf(mn, Ac[row][qd * 16 + c2]);
  red_a[qd][row] = mn;
  __syncthreads();
  mn = fminf(fminf(red_a[0][row], red_a[1][row]),
             fminf(red_a[2][row], red_a[3][row]));
  float s = 0.f;
  float e[16];
#pragma unroll
  for (int c2 = 0; c2 < 16; ++c2) {
    e[c2] = __expf(mn - Ac[row][qd * 16 + c2]);
    s += e[c2];
  }
  red_b[qd][row] = s;
  __syncthreads();
  s = red_b[0][row] + red_b[1][row] + red_b[2][row] + red_b[3][row];
  float inv = 1.f / s;
#pragma unroll
  for (int c2 = 0; c2 < 16; ++c2)
    Asoft[(size_t)b * CH * CH + row * CH + qd * 16 + c2] = e[c2] * inv;
}

// ---------------------------------------------------------------------------
// Kernel 5: fused = c_gamma * (Asoft @ c_x) + c_x + samap.  grid = (HW/64, B).
// ---------------------------------------------------------------------------
__global__ __launch_bounds__(256)
void chan_apply_fuse(const float* __restrict__ cx, const float* __restrict__ Asoft,
                     const float* __restrict__ samap, const float* __restrict__ cg,
                     float* __restrict__ fused) {
  const int p0 = blockIdx.x * 64;
  const int b  = blockIdx.y;
  __shared__ float As[CH][CH + 2];
  __shared__ float Xt[CH][64 + 1];

  for (int idx = threadIdx.x; idx < CH * 64; idx += 256) {
    int i2 = idx >> 6, j2 = idx & 63;
    As[i2][j2] = Asoft[(size_t)b * CH * CH + i2 * CH + j2];
    Xt[i2][j2] = cx[((size_t)b * CH + i2) * HWP + p0 + j2];
  }
  __syncthreads();

  const int lane = threadIdx.x & 31, wv = threadIdx.x >> 5;
  const int lo = lane & 15, hi = lane >> 4;
  const int nt = wv & 3, m0 = (wv >> 2) * 32;

  v8f acc0 = {}, acc1 = {};
  for (int ks = 0; ks < CH; ks += 4) {
    v2f a0, a1, bf;
    bf[0] = Xt[ks + hi * 2 + 0][nt * 16 + lo];
    bf[1] = Xt[ks + hi * 2 + 1][nt * 16 + lo];
    a0[0] = As[m0 + lo][ks + hi * 2 + 0];
    a0[1] = As[m0 + lo][ks + hi * 2 + 1];
    a1[0] = As[m0 + 16 + lo][ks + hi * 2 + 0];
    a1[1] = As[m0 + 16 + lo][ks + hi * 2 + 1];
    acc0 = wmma_f32(a0, bf, acc0);
    acc1 = wmma_f32(a1, bf, acc1);
  }
  const float g = cg[0];
  const int n = nt * 16 + lo;
#pragma unroll
  for (int r = 0; r < 8; ++r) {
    int o0 = m0 + r + hi * 8;
    int o1 = o0 + 16;
    fused[((size_t)b * CH + o0) * HWP + p0 + n] =
        g * acc0[r] + Xt[o0][n] + samap[((size_t)b * CH + o0) * HWP + p0 + n];
    fused[((size_t)b * CH + o1) * HWP + p0 + n] =
        g * acc1[r] + Xt[o1][n] + samap[((size_t)b * CH + o1) * HWP + p0 + n];
  }
}

// ---------------------------------------------------------------------------
extern "C" void kernel_launch(void* const* d_in, const int* in_sizes, int n_in,
                              void* d_out, int out_size, void* d_ws, size_t ws_size,
                              hipStream_t stream) {
  const float* x    = (const float*)d_in[0];
  const float* sW   = (const float*)d_in[1];
  const float* sb   = (const float*)d_in[2];
  const float* s_g  = (const float*)d_in[3];
  const float* s_b  = (const float*)d_in[4];
  const float* s_m  = (const float*)d_in[5];
  const float* s_v  = (const float*)d_in[6];
  const float* cW   = (const float*)d_in[7];
  const float* cb   = (const float*)d_in[8];
  const float* c_g  = (const float*)d_in[9];
  const float* c_b  = (const float*)d_in[10];
  const float* c_m  = (const float*)d_in[11];
  const float* c_v  = (const float*)d_in[12];
  const float* qW   = (const float*)d_in[13];
  const float* qb   = (const float*)d_in[14];
  const float* kW   = (const float*)d_in[15];
  const float* kb   = (const float*)d_in[16];
  const float* vW   = (const float*)d_in[17];
  const float* vb   = (const float*)d_in[18];
  const float* oW   = (const float*)d_in[19];
  const float* ob   = (const float*)d_in[20];
  const float* s_gm = (const float*)d_in[21];
  const float* c_gm = (const float*)d_in[22];

  // workspace carve-up (floats)
  float* ws    = (float*)d_ws;
  float* sx    = ws;                    // 2*64*4096 = 524288
  float* cxb   = sx    + 524288;        // 524288
  float* qbuf  = cxb   + 524288;        // 2*8*4096  = 65536
  float* kbuf  = qbuf  + 65536;         // 65536
  float* vbuf  = kbuf  + 65536;         // 524288
  float* samap = vbuf  + 524288;        // 524288
  float* Asoft = samap + 524288;        // 2*64*64   = 8192
  float* fused = Asoft + 8192;          // 524288

  const dim3 blk(256);
  const dim3 gRow(HT, 2);      // (H, B)
  const dim3 gPix(HWP / 64, 2);

  conv3x3_bn_relu<<<gRow, blk, 0, stream>>>(x, sW, sb, s_g, s_b, s_m, s_v, sx);
  conv3x3_bn_relu<<<gRow, blk, 0, stream>>>(x, cW, cb, c_g, c_b, c_m, c_v, cxb);

  conv1x1<<<gPix, blk, 0, stream>>>(x, qW, qb, qbuf, 8);
  conv1x1<<<gPix, blk, 0, stream>>>(x, kW, kb, kbuf, 8);
  conv1x1<<<gPix, blk, 0, stream>>>(x, vW, vb, vbuf, 64);

  spatial_attn<<<gPix, blk, 0, stream>>>(qbuf, kbuf, vbuf, sx, s_gm, samap);

  chan_gram_softmax<<<dim3(2), blk, 0, stream>>>(cxb, Asoft);
  chan_apply_fuse<<<gPix, blk, 0, stream>>>(cxb, Asoft, samap, c_gm, fused);

  conv1x1<<<gPix, blk, 0, stream>>>(fused, oW, ob, (float*)d_out, 64);
}